// Encoder2_18734647345438
// MI455X (gfx1250) — compile-verified
//
#include <hip/hip_runtime.h>

#define HDIM 128
#define LNEPS 1e-5f

typedef __attribute__((ext_vector_type(2))) float v2f;
typedef __attribute__((ext_vector_type(8))) float v8f;

// Hardware no-return fp32 atomic add (STOREcnt-tracked, resolves in L2).
__device__ __forceinline__ void atomic_add_f32_noret(float* p, float v) {
  asm volatile("global_atomic_add_f32 %0, %1, off" :: "v"(p), "v"(v) : "memory");
}

// ---------------- init: agg = 0, deg = 1 (self loop) ----------------
__global__ __launch_bounds__(256) void k_init(float* __restrict__ deg,
                                              float* __restrict__ agg, int n) {
  int i = blockIdx.x * 256 + threadIdx.x;
  if (i < n * HDIM) agg[i] = 0.0f;
  if (i < n) deg[i] = 1.0f;
}

// ---------------- degree count over target nodes ----------------
__global__ __launch_bounds__(256) void k_deg(const int* __restrict__ col,
                                             float* __restrict__ deg, int e) {
  int i = blockIdx.x * 256 + threadIdx.x;
  if (i < e) atomic_add_f32_noret(&deg[col[i]], 1.0f);
}

// ---------------- deg -> dinv (deg >= 1 always due to self loop) ----------------
__global__ __launch_bounds__(256) void k_dinv(float* __restrict__ deg, int n) {
  int i = blockIdx.x * 256 + threadIdx.x;
  if (i < n) deg[i] = rsqrtf(deg[i]);
}

// ---------------- xw = x @ W via V_WMMA_F32_16X16X4_F32 ----------------
// Block: 256 threads = 8 waves. Block computes 16 rows x 128 cols of xw.
// Wave w owns output tile [16 x 16] at columns [16w, 16w+16).
__global__ __launch_bounds__(256) void k_gemm(const float* __restrict__ x,
                                              const float* __restrict__ W,
                                              float* __restrict__ xw, int n) {
  __shared__ float xs[16][HDIM + 4];  // +4 pad: kills 16-way LDS bank conflicts
  const int row0 = blockIdx.x * 16;

  // Cooperative load of the 16x128 x-tile (8 floats per thread).
  for (int i = threadIdx.x; i < 16 * HDIM; i += 256) {
    int r = i >> 7, c = i & (HDIM - 1);
    int gr = row0 + r;
    xs[r][c] = (gr < n) ? x[(size_t)gr * HDIM + c] : 0.0f;
  }
  __syncthreads();

  const int wave = threadIdx.x >> 5;   // 0..7
  const int lane = threadIdx.x & 31;
  const int m    = lane & 15;          // row within tile
  const int half = lane >> 4;          // 0: K={0,1}; 1: K={2,3}
  const int ncol = (wave << 4) + m;    // output column

  v8f acc = {};
#pragma unroll
  for (int k = 0; k < HDIM; k += 4) {
    v2f a, bfrag;
    // A (16x4 fp32): lanes 0-15 hold K=0 (v0) / K=1 (v1); lanes 16-31 K=2 / K=3
    a.x = xs[m][k + 2 * half];
    a.y = xs[m][k + 2 * half + 1];
    // B (4x16 fp32): row-striped across lanes, mirrored K split
    bfrag.x = W[(size_t)(k + 2 * half) * HDIM + ncol];
    bfrag.y = W[(size_t)(k + 2 * half + 1) * HDIM + ncol];
    acc = __builtin_amdgcn_wmma_f32_16x16x4_f32(
        /*neg_a=*/false, a, /*neg_b=*/false, bfrag,
        /*c_mod=*/(short)0, acc, /*reuse_a=*/false, /*reuse_b=*/false);
  }

  // C/D layout: VGPR r -> (M = r + 8*half, N = lane&15)
#pragma unroll
  for (int r = 0; r < 8; ++r) {
    int gr = row0 + r + 8 * half;
    if (gr < n) xw[(size_t)gr * HDIM + ncol] = acc[r];
  }
}

// ---------------- edge scatter: agg[col] += norm * xw[row] ----------------
// One wave per edge; each lane handles a float4 (32 lanes * 4 = 128 features).
__global__ __launch_bounds__(256) void k_scatter(const float* __restrict__ xw,
                                                 const int* __restrict__ row,
                                                 const int* __restrict__ col,
                                                 const float* __restrict__ dinv,
                                                 float* __restrict__ agg, int e) {
  int eid = blockIdx.x * 8 + (threadIdx.x >> 5);
  if (eid >= e) return;
  int lane = threadIdx.x & 31;
  int r = row[eid];
  int c = col[eid];
  float nrm = dinv[r] * dinv[c];
  float4 v = ((const float4*)(xw + (size_t)r * HDIM))[lane];
  float* dst = agg + (size_t)c * HDIM + lane * 4;
  atomic_add_f32_noret(dst + 0, v.x * nrm);
  atomic_add_f32_noret(dst + 1, v.y * nrm);
  atomic_add_f32_noret(dst + 2, v.z * nrm);
  atomic_add_f32_noret(dst + 3, v.w * nrm);
}

// ---------------- fused self-loop + bias + residual + LayerNorm ----------------
// One wave per node; lane holds 4 features; wave32 shuffle reductions.
__global__ __launch_bounds__(256) void k_finalize(const float* __restrict__ x,
                                                  const float* __restrict__ xw,
                                                  const float* __restrict__ agg,
                                                  const float* __restrict__ dinv,
                                                  const float* __restrict__ bias,
                                                  float* __restrict__ out, int n) {
  int node = blockIdx.x * 8 + (threadIdx.x >> 5);
  if (node >= n) return;
  int lane = threadIdx.x & 31;
  size_t base = (size_t)node * HDIM + lane * 4;

  float di = dinv[node];
  float selfw = di * di;  // self-loop norm = dinv[i]*dinv[i]

  float4 a4 = *(const float4*)(agg + base);
  float4 w4 = *(const float4*)(xw + base);
  float4 x4 = *(const float4*)(x + base);
  float4 b4 = *(const float4*)(bias + lane * 4);

  float h0 = a4.x + selfw * w4.x + b4.x + x4.x;
  float h1 = a4.y + selfw * w4.y + b4.y + x4.y;
  float h2 = a4.z + selfw * w4.z + b4.z + x4.z;
  float h3 = a4.w + selfw * w4.w + b4.w + x4.w;

  float s = h0 + h1 + h2 + h3;
#pragma unroll
  for (int off = 16; off > 0; off >>= 1) s += __shfl_xor(s, off, 32);
  float mu = s * (1.0f / HDIM);

  float d0 = h0 - mu, d1 = h1 - mu, d2 = h2 - mu, d3 = h3 - mu;
  float vs = d0 * d0 + d1 * d1 + d2 * d2 + d3 * d3;
#pragma unroll
  for (int off = 16; off > 0; off >>= 1) vs += __shfl_xor(vs, off, 32);
  float inv = rsqrtf(vs * (1.0f / HDIM) + LNEPS);

  float4 o;
  o.x = d0 * inv; o.y = d1 * inv; o.z = d2 * inv; o.w = d3 * inv;
  *(float4*)(out + base) = o;
}

extern "C" void kernel_launch(void* const* d_in, const int* in_sizes, int n_in,
                              void* d_out, int out_size, void* d_ws, size_t ws_size,
                              hipStream_t stream) {
  const float* x   = (const float*)d_in[0];
  const int*   ei  = (const int*)d_in[1];   // [2, E]: rows then cols
  const float* W   = (const float*)d_in[3];
  const float* b   = (const float*)d_in[4];
  float*       out = (float*)d_out;

  const int n = in_sizes[0] / HDIM;
  const int e = in_sizes[1] / 2;
  const int* rows = ei;
  const int* cols = ei + e;

  // Workspace layout: deg/dinv [npad] | xw [n*H] | agg [n*H]
  size_t npad = ((size_t)n + 255) & ~(size_t)255;
  float* deg = (float*)d_ws;
  float* xw  = deg + npad;
  float* agg = xw + (size_t)n * HDIM;

  k_init<<<(int)(((size_t)n * HDIM + 255) / 256), 256, 0, stream>>>(deg, agg, n);
  k_deg<<<(e + 255) / 256, 256, 0, stream>>>(cols, deg, e);
  k_dinv<<<(n + 255) / 256, 256, 0, stream>>>(deg, n);
  k_gemm<<<(n + 15) / 16, 256, 0, stream>>>(x, W, xw, n);
  k_scatter<<<(e + 7) / 8, 256, 0, stream>>>(xw, rows, cols, deg, agg, e);
  k_finalize<<<(n + 7) / 8, 256, 0, stream>>>(x, xw, agg, deg, b, out, n);
}